// TrajGPT_53506702573937
// MI455X (gfx1250) — compile-verified
//
#include <hip/hip_runtime.h>

// ---------------------------------------------------------------------------
// Problem sizes (fixed by the reference)
// ---------------------------------------------------------------------------
#define B_ROWS  4096
#define G_GENES 20000
#define H_DIM   64
#define L_LAYERS 3
#define KSPLIT  10          // K-split chunks for the input projection
#define KCHUNK  (G_GENES / KSPLIT)   // 2000, multiple of 4

typedef __attribute__((ext_vector_type(2))) float v2f;
typedef __attribute__((ext_vector_type(8))) float v8f;

__device__ __forceinline__ v8f vzero8() {
  v8f z = {0.f, 0.f, 0.f, 0.f, 0.f, 0.f, 0.f, 0.f};
  return z;
}

// D = A(16x4,f32) * B(4x16,f32) + C(16x16,f32)   -> v_wmma_f32_16x16x4_f32
__device__ __forceinline__ v8f wmma4(v2f a, v2f b, v8f c) {
  return __builtin_amdgcn_wmma_f32_16x16x4_f32(
      /*neg_a=*/false, a, /*neg_b=*/false, b,
      /*c_mod=*/(short)0, c, /*reuse_a=*/false, /*reuse_b=*/false);
}

// ---------------------------------------------------------------------------
// Kernel 1: partial[c][m][n] = sum over K-chunk c of x[m][k] * in_W[n][k]
// 4 waves / WG; wave = 16 rows x 64 cols; grid = (4096/64, KSPLIT)
// ---------------------------------------------------------------------------
__global__ __launch_bounds__(128) void k_inproj(const float* __restrict__ x,
                                                const float* __restrict__ inW,
                                                float* __restrict__ part) {
  const int lane = threadIdx.x & 31;
  const int wave = threadIdx.x >> 5;
  const int hi = lane >> 4;      // lanes 16-31 hold K+2,K+3
  const int ln = lane & 15;      // M (for A) / N (for B) within tile
  const int m0 = blockIdx.x * 64 + wave * 16;
  const int kBeg = blockIdx.y * KCHUNK;

  const float* __restrict__ xrow = x + (size_t)(m0 + ln) * G_GENES + 2 * hi;
  const float* __restrict__ w0 = inW + (size_t)(0 * 16 + ln) * G_GENES + 2 * hi;
  const float* __restrict__ w1 = inW + (size_t)(1 * 16 + ln) * G_GENES + 2 * hi;
  const float* __restrict__ w2 = inW + (size_t)(2 * 16 + ln) * G_GENES + 2 * hi;
  const float* __restrict__ w3 = inW + (size_t)(3 * 16 + ln) * G_GENES + 2 * hi;

  v8f acc0 = vzero8(), acc1 = vzero8(), acc2 = vzero8(), acc3 = vzero8();

  // Outer blocks of 200 floats (KCHUNK = 10 * 200), prefetch the x stream.
  for (int kb = kBeg; kb < kBeg + KCHUNK; kb += 200) {
    __builtin_prefetch(xrow + kb + 800, 0, 1);   // global_prefetch_b8
#pragma unroll 5
    for (int k = kb; k < kb + 200; k += 4) {
      v2f a = *(const v2f*)(xrow + k);
      v2f b0 = *(const v2f*)(w0 + k);
      v2f b1 = *(const v2f*)(w1 + k);
      v2f b2 = *(const v2f*)(w2 + k);
      v2f b3 = *(const v2f*)(w3 + k);
      acc0 = wmma4(a, b0, acc0);
      acc1 = wmma4(a, b1, acc1);
      acc2 = wmma4(a, b2, acc2);
      acc3 = wmma4(a, b3, acc3);
    }
  }

  // C layout: VGPR r -> row (r + 8*hi), col = ln (per 16-wide N tile)
  float* __restrict__ p = part + (size_t)blockIdx.y * B_ROWS * H_DIM;
  const int rowb = m0 + 8 * hi;
#pragma unroll
  for (int r = 0; r < 8; ++r) {
    float* pr = p + (size_t)(rowb + r) * H_DIM + ln;
    pr[0]  = acc0[r];
    pr[16] = acc1[r];
    pr[32] = acc2[r];
    pr[48] = acc3[r];
  }
}

// ---------------------------------------------------------------------------
// Kernel 2: h[m][n] = sum_c partial[c][m][n] + in_b[n]
// ---------------------------------------------------------------------------
__global__ __launch_bounds__(256) void k_reduce(const float* __restrict__ part,
                                                const float* __restrict__ in_b,
                                                float* __restrict__ h) {
  const int idx = blockIdx.x * 256 + threadIdx.x;  // 0 .. 4096*64-1
  float s = in_b[idx & (H_DIM - 1)];
#pragma unroll
  for (int c = 0; c < KSPLIT; ++c)
    s += part[(size_t)c * B_ROWS * H_DIM + idx];
  h[idx] = s;
}

// ---------------------------------------------------------------------------
// LDS-sourced WMMA GEMM helper for the fused middle kernel.
// D(32 x 16*NCT*2... ) : 4 waves, wave (w&1) picks 16-row block,
// wave (w>>1) picks half of the NCT*2 column tiles; NCT tiles per wave.
// W is [ncols x K] row-major (i.e. we compute A @ W^T), bias per column.
// ---------------------------------------------------------------------------
template <int NCT, bool RELU>
__device__ __forceinline__ void gemm_ws(const float* __restrict__ sA, int strideA,
                                        int K,
                                        const float* __restrict__ W,
                                        const float* __restrict__ bias,
                                        float* __restrict__ sD, int strideD) {
  const int lane = threadIdx.x & 31;
  const int wave = threadIdx.x >> 5;
  const int hi = lane >> 4;
  const int ln = lane & 15;
  const int mblk = (wave & 1) * 16;
  const int ctile0 = (wave >> 1) * NCT;

  v8f acc[NCT];
#pragma unroll
  for (int j = 0; j < NCT; ++j) acc[j] = vzero8();

  for (int k = 0; k < K; k += 4) {
    v2f a = *(const v2f*)(sA + (mblk + ln) * strideA + k + 2 * hi);
#pragma unroll
    for (int j = 0; j < NCT; ++j) {
      const int n = (ctile0 + j) * 16 + ln;
      v2f b = *(const v2f*)(W + (size_t)n * K + k + 2 * hi);
      acc[j] = wmma4(a, b, acc[j]);
    }
  }

#pragma unroll
  for (int j = 0; j < NCT; ++j) {
    const int n = (ctile0 + j) * 16 + ln;
    const float bv = bias[n];
#pragma unroll
    for (int r = 0; r < 8; ++r) {
      float v = acc[j][r] + bv;
      if (RELU) v = v > 0.f ? v : 0.f;
      sD[(mblk + r + 8 * hi) * strideD + n] = v;
    }
  }
}

// ---------------------------------------------------------------------------
// Kernel 3: fused 3 layers + FFN, row-independent. 128 thr / 32 rows per WG.
// pool regions (floats, stride 68): H[0], A[2176], B[4352], C[6528]
// FF buffer (stride 136, 32x128) overlays A+B.
// ---------------------------------------------------------------------------
__global__ __launch_bounds__(128) void k_layers(
    float* __restrict__ h,
    const float* __restrict__ qW, const float* __restrict__ qb,
    const float* __restrict__ kW, const float* __restrict__ kb,
    const float* __restrict__ vW, const float* __restrict__ vb,
    const float* __restrict__ oW, const float* __restrict__ ob,
    const float* __restrict__ f1W, const float* __restrict__ f1b,
    const float* __restrict__ f2W, const float* __restrict__ f2b) {
  __shared__ float pool[8704];
  __shared__ float sS[32];
  const int tid = threadIdx.x;
  const int row0 = blockIdx.x * 32;
  const int S = 68;
  float* sH = pool;
  float* sA = pool + 2176;
  float* sB = pool + 4352;
  float* sC = pool + 6528;
  float* sFF = pool + 2176;  // 32 x 128, stride 136 (overlays sA+sB)

  for (int i = tid; i < 32 * 64; i += 128) {
    const int r = i >> 6, c = i & 63;
    sH[r * S + c] = h[(size_t)(row0 + r) * H_DIM + c];
  }
  __syncthreads();

  for (int L = 0; L < L_LAYERS; ++L) {
    const float* qWl = qW + L * H_DIM * H_DIM;
    const float* kWl = kW + L * H_DIM * H_DIM;
    const float* vWl = vW + L * H_DIM * H_DIM;
    const float* oWl = oW + L * H_DIM * H_DIM;
    // q, k, v (independent -> one barrier after all three)
    gemm_ws<2, false>(sH, S, H_DIM, qWl, qb + L * H_DIM, sA, S);
    gemm_ws<2, false>(sH, S, H_DIM, kWl, kb + L * H_DIM, sB, S);
    gemm_ws<2, false>(sH, S, H_DIM, vWl, vb + L * H_DIM, sC, S);
    __syncthreads();
    // s[r] = q[r] . k[r]   (degenerate single-step SRA)
    if (tid < 32) {
      float s = 0.f;
      for (int n = 0; n < H_DIM; ++n) s += sA[tid * S + n] * sB[tid * S + n];
      sS[tid] = s;
    }
    __syncthreads();
    // o = s * v  -> sA
    for (int i = tid; i < 32 * 64; i += 128) {
      const int r = i >> 6, c = i & 63;
      sA[r * S + c] = sS[r] * sC[r * S + c];
    }
    __syncthreads();
    // attn_out = o @ oW^T + ob  -> sB
    gemm_ws<2, false>(sA, S, H_DIM, oWl, ob + L * H_DIM, sB, S);
    __syncthreads();
    // residual
    for (int i = tid; i < 32 * 64; i += 128) {
      const int r = i >> 6, c = i & 63;
      sH[r * S + c] += sB[r * S + c];
    }
    __syncthreads();
  }

  // FFN: relu(h @ f1W^T + f1b) @ f2W^T + f2b, residual, write back
  gemm_ws<4, true>(sH, S, H_DIM, f1W, f1b, sFF, 136);
  __syncthreads();
  gemm_ws<2, false>(sFF, 136, 2 * H_DIM, f2W, f2b, sC, S);
  __syncthreads();
  for (int i = tid; i < 32 * 64; i += 128) {
    const int r = i >> 6, c = i & 63;
    h[(size_t)(row0 + r) * H_DIM + c] = sH[r * S + c] + sC[r * S + c];
  }
}

// ---------------------------------------------------------------------------
// Kernel 4: out[m][n] = sum_k h[m][k] * outW[n][k] + outb[n]
// one 16x16 tile per wave; 4 waves/WG; 256 Mtiles * 1250 Ntiles = 320000 tiles
// ---------------------------------------------------------------------------
__global__ __launch_bounds__(128) void k_outproj(const float* __restrict__ h,
                                                 const float* __restrict__ outW,
                                                 const float* __restrict__ outb,
                                                 float* __restrict__ out) {
  const int lane = threadIdx.x & 31;
  const int wave = threadIdx.x >> 5;
  const int hi = lane >> 4;
  const int ln = lane & 15;
  const int tile = blockIdx.x * 4 + wave;
  const int nt = tile % (G_GENES / 16);   // 0..1249
  const int mt = tile / (G_GENES / 16);   // 0..255
  const int m0 = mt * 16, n0 = nt * 16;

  const float* __restrict__ arow = h + (size_t)(m0 + ln) * H_DIM + 2 * hi;
  const float* __restrict__ brow = outW + (size_t)(n0 + ln) * H_DIM + 2 * hi;

  v8f acc = vzero8();
#pragma unroll
  for (int k = 0; k < H_DIM; k += 4) {
    v2f a = *(const v2f*)(arow + k);
    v2f b = *(const v2f*)(brow + k);
    acc = wmma4(a, b, acc);
  }

  const float bv = outb[n0 + ln];
#pragma unroll
  for (int r = 0; r < 8; ++r)
    out[(size_t)(m0 + r + 8 * hi) * G_GENES + n0 + ln] = acc[r] + bv;
}

// ---------------------------------------------------------------------------
extern "C" void kernel_launch(void* const* d_in, const int* in_sizes, int n_in,
                              void* d_out, int out_size, void* d_ws, size_t ws_size,
                              hipStream_t stream) {
  const float* x    = (const float*)d_in[0];
  const float* in_W = (const float*)d_in[1];
  const float* in_b = (const float*)d_in[2];
  const float* qW   = (const float*)d_in[3];
  const float* qb   = (const float*)d_in[4];
  const float* kW   = (const float*)d_in[5];
  const float* kb   = (const float*)d_in[6];
  const float* vW   = (const float*)d_in[7];
  const float* vb   = (const float*)d_in[8];
  // d_in[9] = dW, d_in[10] = db: provably unused (gamma * S0 with S0 == 0)
  const float* oW   = (const float*)d_in[11];
  const float* ob   = (const float*)d_in[12];
  const float* f1W  = (const float*)d_in[13];
  const float* f1b  = (const float*)d_in[14];
  const float* f2W  = (const float*)d_in[15];
  const float* f2b  = (const float*)d_in[16];
  const float* outW = (const float*)d_in[17];
  const float* outb = (const float*)d_in[18];
  float* out = (float*)d_out;

  // workspace: h [4096*64] then partials [KSPLIT*4096*64]  (~11 MB)
  float* h    = (float*)d_ws;
  float* part = h + (size_t)B_ROWS * H_DIM;

  k_inproj<<<dim3(B_ROWS / 64, KSPLIT), 128, 0, stream>>>(x, in_W, part);
  k_reduce<<<(B_ROWS * H_DIM) / 256, 256, 0, stream>>>(part, in_b, h);
  k_layers<<<B_ROWS / 32, 128, 0, stream>>>(h, qW, qb, kW, kb, vW, vb,
                                            oW, ob, f1W, f1b, f2W, f2b);
  k_outproj<<<(256 * (G_GENES / 16)) / 4, 128, 0, stream>>>(h, outW, outb, out);
}